// GCN_1554778161807
// MI455X (gfx1250) — compile-verified
//
#include <hip/hip_runtime.h>

typedef float v2f __attribute__((ext_vector_type(2)));
typedef float v8f __attribute__((ext_vector_type(8)));
typedef unsigned int u32x4 __attribute__((ext_vector_type(4)));
typedef int i32x4 __attribute__((ext_vector_type(4)));
typedef int i32x8 __attribute__((ext_vector_type(8)));

#define DFEAT 128
#define LDS_STRIDE 136   // 128 floats + 8 floats pad per row (TDM pad reproduces this)

#if __has_builtin(__builtin_amdgcn_tensor_load_to_lds)
#define GCN_USE_TDM 1
#else
#define GCN_USE_TDM 0
#endif

// ---------------------------------------------------------------------------
// H[r, :] = X[r, :] @ W   (fp32 WMMA 16x16x4, K = 128)
// Block = 128 threads = 4 waves; each wave computes a 16x128 strip.
// W (128x128 f32) staged into LDS by the Tensor Data Mover with row padding
// (pad_interval = 128 DWORDs, pad_amount = 8 DWORDs -> 136-float stride).
// ---------------------------------------------------------------------------
__global__ __launch_bounds__(128) void gcn_gemm_wmma(
    const float* __restrict__ X, const float* __restrict__ W,
    float* __restrict__ H, int nrows)
{
    __shared__ float Wl[DFEAT * LDS_STRIDE];

    const int t = threadIdx.x;

#if GCN_USE_TDM
    if (t < 32) {   // one wave issues the TDM descriptor (EXEC is ignored by TDM)
        unsigned long long ga = (unsigned long long)(uintptr_t)W;
        // --- D# group 0: count=1 | lds_addr | global_addr | type=2 ---
        u32x4 g0;
        g0.x = 1u;                                            // count = 1 (valid user desc)
        g0.y = 0u;                                            // lds_addr: sole LDS alloc -> 0
        g0.z = (unsigned)ga;                                  // global_addr[31:0]
        g0.w = (unsigned)((ga >> 32) & 0x01FFFFFFu) | (2u << 30);  // addr[56:32] | type=2
        // --- D# group 1 ---
        i32x8 g1;
        g1[0] = (2 << 16)     // data_size = 2 -> 4 bytes
              | (1 << 20)     // pad_enable
              | (6 << 22)     // pad_interval = 6 -> every 128 DWORDs
              | (7 << 25);    // pad_amount  = 7 -> 8 DWORDs (32 B) pad
        g1[1] = (int)(128u << 16);   // tensor_dim0[15:0] = 128 (atomic_barrier_addr = 0)
        g1[2] = (int)(128u << 16);   // tensor_dim0[31:16]=0 | tensor_dim1[15:0] = 128
        g1[3] = (int)(128u << 16);   // tensor_dim1[31:16]=0 | tile_dim0 = 128
        g1[4] = 128;                 // tile_dim1 = 128, tile_dim2 = 0
        g1[5] = 128;                 // tensor_dim0_stride[31:0] = 128 elements
        g1[6] = 0;                   // stride hi | tensor_dim1_stride lo
        g1[7] = 0;
        i32x4 gz = {0, 0, 0, 0};     // groups 2/3 unused for 2D tensors
#if __clang_major__ >= 23
        i32x8 gz8 = {0, 0, 0, 0, 0, 0, 0, 0};
        __builtin_amdgcn_tensor_load_to_lds(g0, g1, gz, gz, gz8, 0);
#else
        __builtin_amdgcn_tensor_load_to_lds(g0, g1, gz, gz, 0);
#endif
        __builtin_amdgcn_s_wait_tensorcnt(0);
    }
    if (nrows < 0) Wl[t] = 0.0f;     // never taken: keeps LDS "may-be-written"
#else
    {   // fallback: cooperative stage of W; thread t copies row t
        const float4* wsrc = (const float4*)(W + t * DFEAT);
        float*        wrow = Wl + t * LDS_STRIDE;
#pragma unroll
        for (int i = 0; i < DFEAT / 4; ++i) ((float4*)wrow)[i] = wsrc[i];
    }
#endif
    __syncthreads();

    const int wave = t >> 5;
    const int lane = t & 31;
    const int half = lane >> 4;      // 0: K,K+1   1: K+2,K+3
    const int l16  = lane & 15;
    const int rowBase = blockIdx.x * 64 + wave * 16;

    int aRow = rowBase + l16;
    if (aRow >= nrows) aRow = nrows - 1;          // clamp loads; store is masked
    const float* aPtr = X + (size_t)aRow * DFEAT + half * 2;

    v8f acc[8] = {};                               // 8 column tiles of 16x16

    for (int k = 0; k < DFEAT; k += 4) {
        // A fragment: lanes 0-15 -> (M=l16, K=k..k+1); lanes 16-31 -> K=k+2..k+3
        v2f a = *(const v2f*)(aPtr + k);
        const float* brow = Wl + (k + half * 2) * LDS_STRIDE + l16;
#pragma unroll
        for (int nt = 0; nt < 8; ++nt) {
            // B fragment: (K, N=nt*16+l16) in v[0], (K+1, N) in v[1]
            v2f b;
            b.x = brow[nt * 16];
            b.y = brow[LDS_STRIDE + nt * 16];
            acc[nt] = __builtin_amdgcn_wmma_f32_16x16x4_f32(
                false, a, false, b, (short)0, acc[nt], false, false);
        }
    }

    // D layout: VGPR v -> row rowBase+v (lanes 0-15) / rowBase+8+v (lanes 16-31)
    const int outRow0 = rowBase + half * 8;
#pragma unroll
    for (int nt = 0; nt < 8; ++nt) {
#pragma unroll
        for (int v = 0; v < 8; ++v) {
            int r = outRow0 + v;
            if (r < nrows) H[(size_t)r * DFEAT + nt * 16 + l16] = acc[nt][v];
        }
    }
}

// ---------------------------------------------------------------------------
// Degree (with self-loop) -> dinv = rsqrt(deg). Computed once, shared by layers.
// ---------------------------------------------------------------------------
__global__ void deg_init(float* __restrict__ deg, int n)
{
    int i = blockIdx.x * 256 + threadIdx.x;
    if (i < n) deg[i] = 1.0f;                      // self-loop
}

__global__ void deg_accum(const int* __restrict__ dst, float* __restrict__ deg,
                          int e, int n)
{
    int i = blockIdx.x * 256 + threadIdx.x;
    if (i < e) {
        int d = dst[i];
        d = d < 0 ? 0 : (d >= n ? n - 1 : d);
        atomicAdd(deg + d, 1.0f);
    }
}

__global__ void deg_finalize(float* __restrict__ deg, int n)
{
    int i = blockIdx.x * 256 + threadIdx.x;
    if (i < n) deg[i] = rsqrtf(deg[i]);            // deg >= 1 always
}

// ---------------------------------------------------------------------------
// out[n,:] = h[n,:] * dinv[n]^2   (self-loop term == full init of out buffer)
// ---------------------------------------------------------------------------
__global__ void selfloop_init(const float* __restrict__ h,
                              const float* __restrict__ dinv,
                              float* __restrict__ out, int n)
{
    int i = blockIdx.x * 256 + threadIdx.x;
    if (i < n * DFEAT) {
        float s = dinv[i >> 7];
        out[i] = h[i] * s * s;
    }
}

// ---------------------------------------------------------------------------
// out[dst,:] += h[src,:] * dinv[src]*dinv[dst].  One wave32 per edge,
// 4 fp32 atomics per lane (128 columns). h/out live in L2 (51 MB each).
// src/dst are wave-uniform -> broadcast into SGPRs via readfirstlane.
// ---------------------------------------------------------------------------
__global__ __launch_bounds__(256) void edge_aggregate(
    const float* __restrict__ h, const float* __restrict__ dinv,
    const int* __restrict__ src, const int* __restrict__ dst,
    int e, int n, float* __restrict__ out)
{
    int edge = blockIdx.x * 8 + (threadIdx.x >> 5);
    int lane = threadIdx.x & 31;
    if (edge >= e) return;

    int s = src[edge], d = dst[edge];
    s = s < 0 ? 0 : (s >= n ? n - 1 : s);
    d = d < 0 ? 0 : (d >= n ? n - 1 : d);
    s = __builtin_amdgcn_readfirstlane(s);
    d = __builtin_amdgcn_readfirstlane(d);

    float scale = dinv[s] * dinv[d];
    const float* hp = h   + (size_t)s * DFEAT + lane;
    float*       op = out + (size_t)d * DFEAT + lane;
#pragma unroll
    for (int j = 0; j < 4; ++j)
        atomicAdd(op + j * 32, hp[j * 32] * scale);
}

// ---------------------------------------------------------------------------
// out += bias (broadcast over rows), optional ReLU
// ---------------------------------------------------------------------------
__global__ void bias_act(float* __restrict__ out, const float* __restrict__ b,
                         int n, int relu)
{
    int i = blockIdx.x * 256 + threadIdx.x;
    if (i < n * DFEAT) {
        float v = out[i] + b[i & (DFEAT - 1)];
        if (relu) v = fmaxf(v, 0.0f);
        out[i] = v;
    }
}

// ---------------------------------------------------------------------------
extern "C" void kernel_launch(void* const* d_in, const int* in_sizes, int n_in,
                              void* d_out, int out_size, void* d_ws, size_t ws_size,
                              hipStream_t stream)
{
    const float* x  = (const float*)d_in[0];
    const int*   ei = (const int*)d_in[1];     // [2, E]: row 0 = src, row 1 = dst
    const float* W1 = (const float*)d_in[2];
    const float* b1 = (const float*)d_in[3];
    const float* W2 = (const float*)d_in[4];
    const float* b2 = (const float*)d_in[5];

    const int n = in_sizes[0] / DFEAT;
    const int e = in_sizes[1] / 2;
    const int* srcIdx = ei;
    const int* dstIdx = ei + e;

    float* dinv = (float*)d_ws;                          // n floats
    float* bufA = (float*)((char*)d_ws + (1 << 20));     // n*128 floats (51.2 MB)
    float* out  = (float*)d_out;

    const int tpb        = 256;
    const int nodeBlocks = (n + tpb - 1) / tpb;
    const int elemBlocks = (n * DFEAT + tpb - 1) / tpb;
    const int edgeBlocks = (e + tpb - 1) / tpb;
    const int gemmBlocks = (n + 63) / 64;
    const int aggBlocks  = (e + 7) / 8;

    // degree / normalization (shared by both layers)
    deg_init<<<nodeBlocks, tpb, 0, stream>>>(dinv, n);
    deg_accum<<<edgeBlocks, tpb, 0, stream>>>(dstIdx, dinv, e, n);
    deg_finalize<<<nodeBlocks, tpb, 0, stream>>>(dinv, n);

    // ---- layer 1: h1 = x@W1 -> bufA; a1 = relu(agg(h1)+b1) -> d_out ----
    gcn_gemm_wmma<<<gemmBlocks, 128, 0, stream>>>(x, W1, bufA, n);
    selfloop_init<<<elemBlocks, tpb, 0, stream>>>(bufA, dinv, out, n);
    edge_aggregate<<<aggBlocks, tpb, 0, stream>>>(bufA, dinv, srcIdx, dstIdx, e, n, out);
    bias_act<<<elemBlocks, tpb, 0, stream>>>(out, b1, n, 1);

    // ---- layer 2: h2 = a1@W2 -> bufA; out = agg(h2)+b2 -> d_out ----
    gcn_gemm_wmma<<<gemmBlocks, 128, 0, stream>>>(out, W2, bufA, n);
    selfloop_init<<<elemBlocks, tpb, 0, stream>>>(bufA, dinv, out, n);
    edge_aggregate<<<aggBlocks, tpb, 0, stream>>>(bufA, dinv, srcIdx, dstIdx, e, n, out);
    bias_act<<<elemBlocks, tpb, 0, stream>>>(out, b2, n, 0);
}